// HSGNN_28132035789275
// MI455X (gfx1250) — compile-verified
//
#include <hip/hip_runtime.h>
#include <hip/hip_bf16.h>

// ---------------- problem constants ----------------
#define B_    2
#define T_    8
#define N_    512
#define FR_   64
#define FA_   158
#define D_    128
#define NH_   4
#define C_    32
#define TOPK_ 10

typedef float v2f __attribute__((ext_vector_type(2)));
typedef float v8f __attribute__((ext_vector_type(8)));

__device__ __forceinline__ float sigf(float x) { return 1.0f / (1.0f + __expf(-x)); }

__device__ __forceinline__ float wred32(float s) {
#pragma unroll
  for (int off = 16; off >= 1; off >>= 1) s += __shfl_xor(s, off, 32);
  return s;
}

#define WMMA_F32(a, b, c) \
  __builtin_amdgcn_wmma_f32_16x16x4_f32(false, (a), false, (b), (short)0, (c), false, false)

// ======================================================================
// WMMA f32 GEMM:  C[z] = act(A[z] @ W + bias)
// A: M x K (row stride lda, batch stride sAz)   W: K x Nn row-major
// REQUIRES: M % 32 == 0, Nn % 32 == 0 (guaranteed by all call sites).
// K may be ragged (single guarded tail step handles K % 4 != 0).
// One wave per 32x32 C tile: 2 A-frags x 2 B-frags -> 4 wmma per K-step.
// A-frag loaded as one b64 (K-consecutive pair per lane).
// ======================================================================
__global__ void k_gemm(const float* __restrict__ A, long long sAz, int lda,
                       const float* __restrict__ W,
                       const float* __restrict__ bias,
                       float* __restrict__ Cm, long long sCz, int ldc,
                       int M, int Nn, int K, int act) {
  const int z = blockIdx.z;
  A  += (long long)z * sAz;
  Cm += (long long)z * sCz;

  const int lane = threadIdx.x & 31;
  const int hseg = lane >> 4;   // lane-half: K-pair select (ISA f32 frag layout)
  const int l16  = lane & 15;
  const int tn   = blockIdx.x * 32;
  const int tm   = blockIdx.y * 32;

  const float* pA0 = A + (long long)(tm + l16) * lda + 2 * hseg;
  const float* pA1 = pA0 + (long long)16 * lda;
  const float* pB  = W + (long long)(2 * hseg) * Nn + tn + l16;

  v8f acc00 = {}, acc01 = {}, acc10 = {}, acc11 = {};
  const int K4 = K & ~3;
#pragma unroll 2
  for (int k0 = 0; k0 < K4; k0 += 4) {
    const v2f a0 = *(const v2f*)pA0;            // one global_load_b64
    const v2f a1 = *(const v2f*)pA1;
    v2f b0, b1;
    b0[0] = pB[0];
    b0[1] = pB[Nn];
    b1[0] = pB[16];
    b1[1] = pB[Nn + 16];
    acc00 = WMMA_F32(a0, b0, acc00);
    acc01 = WMMA_F32(a0, b1, acc01);
    acc10 = WMMA_F32(a1, b0, acc10);
    acc11 = WMMA_F32(a1, b1, acc11);
    pA0 += 4;
    pA1 += 4;
    pB  += (long long)4 * Nn;
  }
  if (K4 < K) {                                  // ragged-K tail (K=158 case)
    v2f a0, a1, b0, b1;
#pragma unroll
    for (int g = 0; g < 2; ++g) {
      const bool ok = (K4 + g + 2 * hseg) < K;
      a0[g] = ok ? pA0[g] : 0.0f;
      a1[g] = ok ? pA1[g] : 0.0f;
      b0[g] = ok ? pB[(long long)g * Nn] : 0.0f;
      b1[g] = ok ? pB[(long long)g * Nn + 16] : 0.0f;
    }
    acc00 = WMMA_F32(a0, b0, acc00);
    acc01 = WMMA_F32(a0, b1, acc01);
    acc10 = WMMA_F32(a1, b0, acc10);
    acc11 = WMMA_F32(a1, b1, acc11);
  }

  const int c0 = tn + l16, c1 = tn + 16 + l16;
  const float bi0 = bias ? bias[c0] : 0.0f;
  const float bi1 = bias ? bias[c1] : 0.0f;
#pragma unroll
  for (int g = 0; g < 8; ++g) {
    const int r0 = tm + g + 8 * hseg;            // C/D: M = g + 8*half
    const int r1 = r0 + 16;
    float v00 = acc00[g] + bi0, v01 = acc01[g] + bi1;
    float v10 = acc10[g] + bi0, v11 = acc11[g] + bi1;
    if (act == 1) {
      v00 = fmaxf(v00, 0.0f); v01 = fmaxf(v01, 0.0f);
      v10 = fmaxf(v10, 0.0f); v11 = fmaxf(v11, 0.0f);
    } else if (act == 2) {
      v00 = sigf(v00); v01 = sigf(v01);
      v10 = sigf(v10); v11 = sigf(v11);
    }
    Cm[(long long)r0 * ldc + c0] = v00;
    Cm[(long long)r0 * ldc + c1] = v01;
    Cm[(long long)r1 * ldc + c0] = v10;
    Cm[(long long)r1 * ldc + c1] = v11;
  }
}

// ======================================================================
// LayerNorm over last dim (128) + optional relu.  One wave per row.
// ======================================================================
__global__ void k_ln(const float* __restrict__ in, const float* __restrict__ g,
                     const float* __restrict__ b, float* __restrict__ out,
                     int rows, int relu) {
  const int row  = blockIdx.x * (blockDim.x >> 5) + (threadIdx.x >> 5);
  const int lane = threadIdx.x & 31;
  if (row >= rows) return;
  const float* x = in + (long long)row * D_;
  float v[4];
  float s = 0.0f;
#pragma unroll
  for (int t = 0; t < 4; ++t) { v[t] = x[lane + 32 * t]; s += v[t]; }
  const float mean = wred32(s) * (1.0f / D_);
  float vs = 0.0f;
#pragma unroll
  for (int t = 0; t < 4; ++t) { const float d = v[t] - mean; vs += d * d; }
  const float inv = rsqrtf(wred32(vs) * (1.0f / D_) + 1e-5f);
#pragma unroll
  for (int t = 0; t < 4; ++t) {
    const int c = lane + 32 * t;
    float y = (v[t] - mean) * inv * g[c] + b[c];
    if (relu) y = fmaxf(y, 0.0f);
    out[(long long)row * D_ + c] = y;
  }
}

// ======================================================================
// Pairwise attention: att[b,i,j] = sigmoid( sum_d relu(ui[d]+uj[d]) * w2[d] + b2 )
// ui already includes attn_b1.  One wave per (b,i); loop over j.
// ======================================================================
__global__ void k_pairwise(const float* __restrict__ ui_all,
                           const float* __restrict__ uj_all,
                           const float* __restrict__ w2,
                           const float* __restrict__ b2p,
                           float* __restrict__ att) {
  const int w = blockIdx.x * (blockDim.x >> 5) + (threadIdx.x >> 5);
  if (w >= B_ * N_) return;
  const int b    = w / N_;
  const int lane = threadIdx.x & 31;
  const float b2 = b2p[0];

  float ui[4], w2r[4];
  const float* uip = ui_all + (long long)w * D_;
#pragma unroll
  for (int t = 0; t < 4; ++t) { ui[t] = uip[lane + 32 * t]; w2r[t] = w2[lane + 32 * t]; }

  const float* ujb = uj_all + (long long)b * N_ * D_;
  float* arow = att + (long long)w * N_;
  for (int j = 0; j < N_; ++j) {
    const float* uj = ujb + (long long)j * D_;
    float s = 0.0f;
#pragma unroll
    for (int t = 0; t < 4; ++t) {
      float p = ui[t] + uj[lane + 32 * t];
      p = fmaxf(p, 0.0f);
      s += p * w2r[t];
    }
    s = wred32(s);
    if (lane == 0) arow[j] = sigf(s + b2);
  }
}

// ======================================================================
// Top-k selection + money-flow thresholding + row normalization.
// One thread per (b,i); emits compact (idx, w_pos, w_neg) of length TOPK.
// ======================================================================
__global__ void k_topk(const float* __restrict__ att, const float* __restrict__ mf,
                       int* __restrict__ tki, float* __restrict__ wpos,
                       float* __restrict__ wneg) {
  const int r = blockIdx.x * blockDim.x + threadIdx.x;
  if (r >= B_ * N_) return;
  const int i = r % N_;
  const float* arow = att + (long long)r * N_;
  const float* mrow = mf  + (long long)r * N_;

  int idx[TOPK_];
#pragma unroll 1
  for (int k = 0; k < TOPK_; ++k) {
    float bv = -1.0f; int bi = 0;
    for (int j = 0; j < N_; ++j) {
      const float v = (j == i) ? 0.0f : arow[j];
      bool taken = false;
      for (int t = 0; t < k; ++t) if (idx[t] == j) taken = true;
      if (!taken && v > bv) { bv = v; bi = j; }
    }
    idx[k] = bi;
  }
  float vp[TOPK_], vn[TOPK_], sp = 0.0f, sn = 0.0f;
#pragma unroll
  for (int k = 0; k < TOPK_; ++k) {
    const int j = idx[k];
    const float a = arow[j], m = mrow[j];
    vp[k] = (m >  0.3f) ? a : 0.0f;
    vn[k] = (m < -0.3f) ? a : 0.0f;
    sp += vp[k]; sn += vn[k];
  }
  const float rp = 1.0f / (sp + 1e-8f), rn = 1.0f / (sn + 1e-8f);
#pragma unroll
  for (int k = 0; k < TOPK_; ++k) {
    tki [(long long)r * TOPK_ + k] = idx[k];
    wpos[(long long)r * TOPK_ + k] = vp[k] * rp;
    wneg[(long long)r * TOPK_ + k] = vn[k] * rn;
  }
}

// ======================================================================
// Sparse message aggregation: out[b,i,:] = sum_k w[k] * proj[b, idx[k], :]
// One wave per (b,i); lanes split D=128.
// ======================================================================
__global__ void k_spmsg(const float* __restrict__ wv, const int* __restrict__ idx,
                        const float* __restrict__ proj, float* __restrict__ out) {
  const int w = blockIdx.x * (blockDim.x >> 5) + (threadIdx.x >> 5);
  if (w >= B_ * N_) return;
  const int b    = w / N_;
  const int lane = threadIdx.x & 31;
  float acc[4] = {0.f, 0.f, 0.f, 0.f};
  const float* pb = proj + (long long)b * N_ * D_;
#pragma unroll
  for (int k = 0; k < TOPK_; ++k) {
    const float wt = wv[(long long)w * TOPK_ + k];
    const float* pr = pb + (long long)idx[(long long)w * TOPK_ + k] * D_;
#pragma unroll
    for (int t = 0; t < 4; ++t) acc[t] += wt * pr[lane + 32 * t];
  }
#pragma unroll
  for (int t = 0; t < 4; ++t) out[(long long)w * D_ + lane + 32 * t] = acc[t];
}

// ======================================================================
// GRU combine.  gi/gh: rows x 3D.  out row stride ldo (writes concat halves).
// ======================================================================
__global__ void k_gru(const float* __restrict__ gi, const float* __restrict__ gh,
                      const float* __restrict__ h, float* __restrict__ out, int ldo) {
  const long long t = (long long)blockIdx.x * blockDim.x + threadIdx.x;
  if (t >= (long long)B_ * N_ * D_) return;
  const int row = (int)(t / D_), d = (int)(t % D_);
  const float* gir = gi + (long long)row * 3 * D_;
  const float* ghr = gh + (long long)row * 3 * D_;
  const float r = sigf(gir[d] + ghr[d]);
  const float z = sigf(gir[d + D_] + ghr[d + D_]);
  const float n = tanhf(gir[d + 2 * D_] + r * ghr[d + 2 * D_]);
  const float hv = h[(long long)row * D_ + d];
  out[(long long)row * ldo + d] = (1.0f - z) * n + z * hv;
}

// ======================================================================
// GAT per-node head scores: os/od[row,h] = dot(xp[row,h,:], a_src/a_dst[h,:])
// ======================================================================
__global__ void k_gatscore(const float* __restrict__ xp, const float* __restrict__ asrc,
                           const float* __restrict__ adst, float* __restrict__ os,
                           float* __restrict__ od) {
  const int t = blockIdx.x * blockDim.x + threadIdx.x;
  if (t >= B_ * N_ * NH_) return;
  const int row = t >> 2, h = t & 3;
  const float* x = xp + (long long)row * D_ + h * C_;
  float ss = 0.0f, sd = 0.0f;
#pragma unroll
  for (int c = 0; c < C_; ++c) { const float v = x[c]; ss += v * asrc[h * C_ + c]; sd += v * adst[h * C_ + c]; }
  os[t] = ss; od[t] = sd;
}

// ======================================================================
// GAT aggregation with online softmax.  One wave per (b,i,h); 32 lanes = 32 chans.
// mask[i,j] = sector[b][j][i] != 0  ||  j == i     (uniform across the wave)
// e = leaky_relu(adst[i,h] + asrc[j,h], 0.2)
// ======================================================================
__global__ void k_gatagg(const float* __restrict__ xp, const float* __restrict__ asrc,
                         const float* __restrict__ adst, const float* __restrict__ adj,
                         const float* __restrict__ bias, float* __restrict__ out) {
  const int w = blockIdx.x * (blockDim.x >> 5) + (threadIdx.x >> 5);
  if (w >= B_ * N_ * NH_) return;
  const int h  = w & 3;
  const int bi = w >> 2;
  const int b  = bi / N_;
  const int i  = bi % N_;
  const int lane = threadIdx.x & 31;

  const float* adjb = adj + (long long)b * N_ * N_;
  const float* xpb  = xp  + (long long)b * N_ * D_;
  const float  di   = adst[(long long)bi * NH_ + h];

  float Mx = -1e30f, S = 0.0f, V = 0.0f;
  for (int j = 0; j < N_; ++j) {
    const bool m = (adjb[(long long)j * N_ + i] != 0.0f) || (j == i);
    if (!m) continue;                              // uniform across wave
    float e = di + asrc[((long long)b * N_ + j) * NH_ + h];
    e = (e >= 0.0f) ? e : 0.2f * e;
    const float xv = xpb[(long long)j * D_ + h * C_ + lane];
    if (e > Mx) {                                  // uniform across wave
      const float sc = __expf(Mx - e);
      S = S * sc + 1.0f;
      V = V * sc + xv;
      Mx = e;
    } else {
      const float wt = __expf(e - Mx);
      S += wt;
      V += wt * xv;
    }
  }
  float o = V / S;
  const int oc = h * C_ + lane;
  if (bias) o += bias[oc];
  out[(long long)bi * D_ + oc] = o;
}

// ---------------- small elementwise fusions ----------------
__global__ void k_elu_add(const float* __restrict__ h0, const float* __restrict__ h1,
                          float* __restrict__ out) {
  const long long t = (long long)blockIdx.x * blockDim.x + threadIdx.x;
  if (t >= (long long)B_ * N_ * D_) return;
  const float a = h1[t];
  out[t] = h0[t] + ((a > 0.0f) ? a : (__expf(a) - 1.0f));
}

__global__ void k_headmean(const float* __restrict__ agg, const float* __restrict__ bias,
                           float* __restrict__ h2) {
  const int t = blockIdx.x * blockDim.x + threadIdx.x;
  if (t >= B_ * N_ * C_) return;
  const int row = t / C_, c = t % C_;
  const float* a = agg + (long long)row * D_;
  h2[t] = 0.25f * (a[c] + a[c + 32] + a[c + 64] + a[c + 96]) + bias[c];
}

__global__ void k_concat2(const float* __restrict__ a, const float* __restrict__ b,
                          float* __restrict__ cat) {
  const long long t = (long long)blockIdx.x * blockDim.x + threadIdx.x;
  if (t >= (long long)B_ * N_ * 2 * D_) return;
  const int row = (int)(t / (2 * D_)), d = (int)(t % (2 * D_));
  cat[t] = (d < D_) ? a[(long long)row * D_ + d] : b[(long long)row * D_ + d - D_];
}

__global__ void k_gatefuse(const float* __restrict__ g, const float* __restrict__ hi,
                           const float* __restrict__ he, float* __restrict__ hf) {
  const long long t = (long long)blockIdx.x * blockDim.x + threadIdx.x;
  if (t >= (long long)B_ * N_ * D_) return;
  const float gv = g[t];
  hf[t] = gv * hi[t] + (1.0f - gv) * he[t];
}

// ======================================================================
// Launcher
// ======================================================================
static inline void gemm(hipStream_t st, const float* A, long long sAz, int lda,
                        const float* W, const float* bias,
                        float* Cm, long long sCz, int ldc,
                        int M, int Nn, int K, int act, int Z) {
  dim3 grid((Nn + 31) / 32, (M + 31) / 32, Z);   // M, Nn multiples of 32 at all call sites
  k_gemm<<<grid, 32, 0, st>>>(A, sAz, lda, W, bias, Cm, sCz, ldc, M, Nn, K, act);
}

extern "C" void kernel_launch(void* const* d_in, const int* in_sizes, int n_in,
                              void* d_out, int out_size, void* d_ws, size_t ws_size,
                              hipStream_t stream) {
  (void)in_sizes; (void)n_in; (void)out_size; (void)ws_size;
  const float* x_risk  = (const float*)d_in[0];
  const float* x_alpha = (const float*)d_in[1];
  const float* mf      = (const float*)d_in[2];
  const float* sector  = (const float*)d_in[3];
  const float* P[44];
  for (int i = 0; i < 44; ++i) P[i] = (const float*)d_in[4 + i];
  // param indices (setup_inputs dict order)
  const float *risk_W = P[0], *risk_b = P[1], *risk_lng = P[2], *risk_lnb = P[3];
  const float *attn_W1 = P[4], *attn_b1 = P[5], *attn_W2 = P[6], *attn_b2 = P[7];
  const float *msgp_W = P[8], *msgp_b = P[9], *msgn_W = P[10], *msgn_b = P[11];
  const float *grup_Wih = P[12], *grup_Whh = P[13], *grup_bih = P[14], *grup_bhh = P[15];
  const float *grun_Wih = P[16], *grun_Whh = P[17], *grun_bih = P[18], *grun_bhh = P[19];
  const float *comb_W = P[20], *comb_b = P[21];
  const float *in_W = P[22], *in_b = P[23], *in_lng = P[24], *in_lnb = P[25];
  const float *gat0_W = P[26], *gat0_as = P[27], *gat0_ad = P[28], *gat0_b = P[29];
  const float *gat1_W = P[30], *gat1_as = P[31], *gat1_ad = P[32], *gat1_b = P[33];
  const float *out_W = P[34], *out_b = P[35];
  const float *gate_W1 = P[36], *gate_b1 = P[37], *gate_W2 = P[38], *gate_b2 = P[39];
  const float *hout_W = P[40], *hout_b = P[41], *hout_lng = P[42], *hout_lnb = P[43];

  // ---- workspace layout (floats) ----
  float* ws = (float*)d_ws;
  const long long BN = (long long)B_ * N_;
  long long o = 0;
  float* H     = ws + o; o += BN * D_;
  float* PRE   = ws + o; o += BN * 3 * D_;    // gi / generic pre-activation
  float* GH    = ws + o; o += BN * 3 * D_;
  float* HW1I  = ws + o; o += BN * D_;
  float* HW1J  = ws + o; o += BN * D_;
  float* ATT   = ws + o; o += BN * N_;
  int*   TKI   = (int*)(ws + o); o += BN * TOPK_;
  float* WPOS  = ws + o; o += BN * TOPK_;
  float* WNEG  = ws + o; o += BN * TOPK_;
  float* MSGP  = ws + o; o += BN * D_;
  float* MBUF  = ws + o; o += BN * D_;
  float* CAT   = ws + o; o += BN * 2 * D_;
  float* HIMP  = ws + o; o += BN * D_;
  float* H0    = ws + o; o += BN * D_;
  float* XP    = ws + o; o += BN * D_;
  float* ASRC  = ws + o; o += BN * NH_;
  float* ADST  = ws + o; o += BN * NH_;
  float* AGG   = ws + o; o += BN * D_;
  float* HA    = ws + o; o += BN * D_;
  float* H2    = ws + o; o += BN * C_;
  float* HEXP  = ws + o; o += BN * D_;
  float* GATE  = ws + o; o += BN * D_;
  float* HF    = ws + o; o += BN * D_;

  const int ED = 256;                         // elementwise block
  const int gBND  = (int)((BN * D_ + ED - 1) / ED);
  const int gWave = (int)(BN / 4);            // 4 waves per 128-thread block

  // ================= implicit branch =================
  // h = relu(LN(x_risk[:,-1] @ risk_W + risk_b))
  gemm(stream, x_risk + (long long)(T_ - 1) * N_ * FR_, (long long)T_ * N_ * FR_, FR_,
       risk_W, risk_b, PRE, (long long)N_ * D_, D_, N_, D_, FR_, 0, B_);
  k_ln<<<(int)(BN / 4), 128, 0, stream>>>(PRE, risk_lng, risk_lnb, H, (int)BN, 1);

  // pairwise attention inputs
  gemm(stream, H, 0, D_, attn_W1,            attn_b1, HW1I, 0, D_, (int)BN, D_, D_, 0, 1);
  gemm(stream, H, 0, D_, attn_W1 + D_ * D_,  nullptr, HW1J, 0, D_, (int)BN, D_, D_, 0, 1);
  k_pairwise<<<gWave, 128, 0, stream>>>(HW1I, HW1J, attn_W2, attn_b2, ATT);
  k_topk<<<(int)((BN + 255) / 256), 256, 0, stream>>>(ATT, mf, TKI, WPOS, WNEG);

  // positive channel
  gemm(stream, H, 0, D_, msgp_W, msgp_b, MSGP, 0, D_, (int)BN, D_, D_, 0, 1);
  k_spmsg<<<gWave, 128, 0, stream>>>(WPOS, TKI, MSGP, MBUF);
  gemm(stream, MBUF, 0, D_, grup_Wih, grup_bih, PRE, 0, 3 * D_, (int)BN, 3 * D_, D_, 0, 1);
  gemm(stream, H,    0, D_, grup_Whh, grup_bhh, GH,  0, 3 * D_, (int)BN, 3 * D_, D_, 0, 1);
  k_gru<<<gBND, ED, 0, stream>>>(PRE, GH, H, CAT, 2 * D_);            // h_pos -> CAT[:, :128]

  // negative channel
  gemm(stream, H, 0, D_, msgn_W, msgn_b, MSGP, 0, D_, (int)BN, D_, D_, 0, 1);
  k_spmsg<<<gWave, 128, 0, stream>>>(WNEG, TKI, MSGP, MBUF);
  gemm(stream, MBUF, 0, D_, grun_Wih, grun_bih, PRE, 0, 3 * D_, (int)BN, 3 * D_, D_, 0, 1);
  gemm(stream, H,    0, D_, grun_Whh, grun_bhh, GH,  0, 3 * D_, (int)BN, 3 * D_, D_, 0, 1);
  k_gru<<<gBND, ED, 0, stream>>>(PRE, GH, H, CAT + D_, 2 * D_);       // h_neg -> CAT[:, 128:]

  // h_imp = [h_pos|h_neg] @ comb_W + comb_b
  gemm(stream, CAT, 0, 2 * D_, comb_W, comb_b, HIMP, 0, D_, (int)BN, D_, 2 * D_, 0, 1);

  // ================= explicit branch =================
  gemm(stream, x_alpha + (long long)(T_ - 1) * N_ * FA_, (long long)T_ * N_ * FA_, FA_,
       in_W, in_b, PRE, (long long)N_ * D_, D_, N_, D_, FA_, 0, B_);
  k_ln<<<(int)(BN / 4), 128, 0, stream>>>(PRE, in_lng, in_lnb, H0, (int)BN, 1);

  // GAT layer 0 (concat)
  gemm(stream, H0, 0, D_, gat0_W, nullptr, XP, 0, D_, (int)BN, D_, D_, 0, 1);
  k_gatscore<<<(int)((BN * NH_ + 255) / 256), 256, 0, stream>>>(XP, gat0_as, gat0_ad, ASRC, ADST);
  k_gatagg<<<(int)(BN * NH_ / 4), 128, 0, stream>>>(XP, ASRC, ADST, sector, gat0_b, AGG);
  k_elu_add<<<gBND, ED, 0, stream>>>(H0, AGG, HA);

  // GAT layer 1 (mean over heads)
  gemm(stream, HA, 0, D_, gat1_W, nullptr, XP, 0, D_, (int)BN, D_, D_, 0, 1);
  k_gatscore<<<(int)((BN * NH_ + 255) / 256), 256, 0, stream>>>(XP, gat1_as, gat1_ad, ASRC, ADST);
  k_gatagg<<<(int)(BN * NH_ / 4), 128, 0, stream>>>(XP, ASRC, ADST, sector, nullptr, AGG);
  k_headmean<<<(int)((BN * C_ + 255) / 256), 256, 0, stream>>>(AGG, gat1_b, H2);
  gemm(stream, H2, 0, C_, out_W, out_b, HEXP, 0, D_, (int)BN, D_, C_, 0, 1);

  // ================= gated fusion + output =================
  k_concat2<<<(int)((BN * 2 * D_ + ED - 1) / ED), ED, 0, stream>>>(HIMP, HEXP, CAT);
  gemm(stream, CAT, 0, 2 * D_, gate_W1, gate_b1, PRE,  0, D_, (int)BN, D_, 2 * D_, 1, 1);
  gemm(stream, PRE, 0, D_,     gate_W2, gate_b2, GATE, 0, D_, (int)BN, D_, D_,     2, 1);
  k_gatefuse<<<gBND, ED, 0, stream>>>(GATE, HIMP, HEXP, HF);
  gemm(stream, HF, 0, D_, hout_W, hout_b, PRE, 0, D_, (int)BN, D_, D_, 0, 1);
  k_ln<<<(int)(BN / 4), 128, 0, stream>>>(PRE, hout_lng, hout_lnb, (float*)d_out, (int)BN, 1);
}